// FHME_84705345011962
// MI455X (gfx1250) — compile-verified
//
#include <hip/hip_runtime.h>
#include <hip/hip_bf16.h>
#include <math.h>

typedef __attribute__((ext_vector_type(2))) float v2f;
typedef __attribute__((ext_vector_type(8))) float v8f;

constexpr int kB = 4, kS = 1024, kD = 256, kH = 8, kNK = 128;
constexpr int kT = kB * kS;                 // 4096 tokens
constexpr int kTopK = 32, kNN = 15;
constexpr int kM = kH * kNK;                // 1024 usable experts

// ---- CDNA5 async global->LDS path (guarded; falls back to plain LDS store) --
#if defined(__has_builtin)
#if __has_builtin(__builtin_amdgcn_global_load_async_to_lds_b32)
#define ATHENA_ASYNC_LDS 1
#endif
#endif

typedef __attribute__((address_space(1))) int g_int;
typedef __attribute__((address_space(3))) int l_int;

__device__ __forceinline__ void async_copy_b32(float* lds_dst,
                                               const float* gsrc) {
#if defined(ATHENA_ASYNC_LDS)
  __builtin_amdgcn_global_load_async_to_lds_b32(
      (g_int*)(gsrc), (l_int*)(lds_dst), /*offset=*/0, /*cpol=*/0);
#else
  *lds_dst = *gsrc;
#endif
}

__device__ __forceinline__ void async_wait_all() {
#if defined(ATHENA_ASYNC_LDS)
#if __has_builtin(__builtin_amdgcn_s_wait_asynccnt)
  __builtin_amdgcn_s_wait_asynccnt(0);
#else
  asm volatile("s_wait_asynccnt 0x0" ::: "memory");
#endif
#endif
}

// ---------------------------------------------------------------------------
// K1: scores_nat[b,h,s,k] = sum_{d<32} x[b,s,h*32+d] * keys[h,k,d]
// One wave computes one 16(s) x 16(k) tile via 8x V_WMMA_F32_16X16X4_F32.
// ---------------------------------------------------------------------------
__global__ __launch_bounds__(32) void k_score_gemm(
    const float* __restrict__ x, const float* __restrict__ keys,
    float* __restrict__ scores) {
  const int lane = threadIdx.x;
  const int s0 = blockIdx.x * 16;
  const int k0 = blockIdx.y * 16;
  const int bh = blockIdx.z;                // b*8 + h
  const int h  = bh & 7;
  const int mr = lane & 15;                 // A row (M) and B col (N)
  const int kb = (lane >> 4) * 2;           // K pair base: 0 or 2 (ISA 16x4 A layout)
  const float* __restrict__ arow =
      x + (size_t)((bh >> 3) * kS + s0 + mr) * kD + h * 32;
  const float* __restrict__ brow =
      keys + (size_t)(h * kNK + k0 + mr) * 32;
  v8f acc = {};
#pragma unroll
  for (int d0 = 0; d0 < 32; d0 += 4) {
    v2f a, b;
    a.x = arow[d0 + kb]; a.y = arow[d0 + kb + 1];
    b.x = brow[d0 + kb]; b.y = brow[d0 + kb + 1];
    acc = __builtin_amdgcn_wmma_f32_16x16x4_f32(false, a, false, b,
                                                (short)0, acc, false, false);
  }
  // D layout: VGPR r <-> M = r + 8*(lane>=16), N = lane&15
  const int half = (lane >> 4) * 8;
  float* __restrict__ dst =
      scores + ((size_t)bh * kS + s0 + half) * kNK + k0 + mr;
#pragma unroll
  for (int r = 0; r < 8; ++r) dst[(size_t)r * kNK] = acc[r];
}

// ---------------------------------------------------------------------------
// K2: per-token scrambled-view gather, top-32 selection, softmax, tree coeffs.
// One wave per token; 1024 candidate scores staged in LDS via async copies.
// ---------------------------------------------------------------------------
__global__ __launch_bounds__(32) void k_topk(
    const float* __restrict__ scores, const unsigned char* __restrict__ bits,
    int* __restrict__ eidx, float* __restrict__ acoef) {
  __shared__ float sc[kM];
  const int t = blockIdx.x;
  const int lane = threadIdx.x;
  const int b  = t >> 10;
  const int sp = t & 1023;                  // s'
  const int h = sp >> 7;                    // torch .view scramble
  const int sbase = (sp & 127) * 8;
  const float* __restrict__ base = scores + (size_t)(b * kH + h) * kS * kNK;
  for (int m = lane; m < kM; m += 32) {
    const int sorig = sbase + (m >> 7);
    const int k = m & 127;
    async_copy_b32(&sc[m], &base[(size_t)sorig * kNK + k]);
  }
  async_wait_all();
  __syncthreads();

  float myv = -INFINITY; int myi = 0;       // lane j keeps j-th largest
  for (int it = 0; it < kTopK; ++it) {
    float bv = -INFINITY; int bi = 0;
    for (int m = lane; m < kM; m += 32) {
      const float v = sc[m];
      if (v > bv) { bv = v; bi = m; }
    }
#pragma unroll
    for (int off = 16; off > 0; off >>= 1) {
      const float ov = __shfl_xor(bv, off, 32);
      const int   oi = __shfl_xor(bi, off, 32);
      if (ov > bv || (ov == bv && oi < bi)) { bv = ov; bi = oi; }
    }
    if (lane == it) { myv = bv; myi = bi; }
    if (lane == 0) sc[bi] = -INFINITY;
    __syncthreads();
  }

  // softmax over the 32 selected values (lane 0 holds the max)
  const float mx = __shfl(myv, 0, 32);
  const float e = expf(myv - mx);
  float s = e;
#pragma unroll
  for (int off = 16; off > 0; off >>= 1) s += __shfl_xor(s, off, 32);
  const float w = e / s;

  // fractal tree coefficients (7 internal nodes, 8 leaves)
  float coeff[kNN], contrib[kNN];
  coeff[0] = 1.f;
  const unsigned char* __restrict__ bp = bits + ((size_t)t * kTopK + lane) * 7;
#pragma unroll
  for (int n = 0; n < 7; ++n) {
    const bool bb = bp[n] != 0;
    const float c = coeff[n];
    contrib[n] = bb ? 0.f : c;
    const float half = bb ? c * 0.5f : 0.f;
    coeff[2 * n + 1] = half;
    coeff[2 * n + 2] = half;
  }
#pragma unroll
  for (int n = 7; n < kNN; ++n) contrib[n] = coeff[n];

  eidx[(size_t)t * kTopK + lane] = myi;
  float* __restrict__ ap = acoef + ((size_t)t * kTopK + lane) * kNN;
#pragma unroll
  for (int n = 0; n < kNN; ++n) ap[n] = w * contrib[n];
}

// ---------------------------------------------------------------------------
// K3: sparse expert accumulate. 256 threads per token (8 waves x 4 slots).
// y[t,d] = Sc_t + sum_{j,n} a[t,j,n] * expert_b[e_j,n,d]
// Sc_t   =        sum_{j,n} a[t,j,n] * dot(x_t, expert_w[e_j,n,:])
// Expert tables (31.5 MB live) are L2-resident -> direct gather loads +
// global_prefetch; x_t staged to LDS via the async path.
// ---------------------------------------------------------------------------
__global__ __launch_bounds__(256) void k_experts(
    const float* __restrict__ x, const float* __restrict__ ew,
    const float* __restrict__ eb, const int* __restrict__ eidx,
    const float* __restrict__ acoef, float* __restrict__ y) {
  __shared__ float xs[kD];
  __shared__ float yacc[kD];
  __shared__ float scacc;
  const int t = blockIdx.x;
  const int tid = threadIdx.x;
  async_copy_b32(&xs[tid], &x[(size_t)t * kD + tid]);
  yacc[tid] = 0.f;
  if (tid == 0) scacc = 0.f;
  async_wait_all();
  __syncthreads();

  const int wave = tid >> 5, lane = tid & 31;
  float bpart[8] = {0.f, 0.f, 0.f, 0.f, 0.f, 0.f, 0.f, 0.f};
  float scp = 0.f;
  for (int j = wave; j < kTopK; j += 8) {
    const int e = eidx[(size_t)t * kTopK + j];
    const float* __restrict__ W  = ew + (size_t)e * kNN * kD;
    const float* __restrict__ Bb = eb + (size_t)e * kNN * kD;
    const float* __restrict__ av = acoef + ((size_t)t * kTopK + j) * kNN;
    __builtin_prefetch(W, 0, 1);            // global_prefetch_b8
    __builtin_prefetch(Bb, 0, 1);
#pragma unroll
    for (int n = 0; n < kNN; ++n) {
      const float an = av[n];
#pragma unroll
      for (int q = 0; q < 8; ++q) {
        const int dd = lane + q * 32;
        scp += an * W[n * kD + dd] * xs[dd];
        bpart[q] += an * Bb[n * kD + dd];
      }
    }
  }
#pragma unroll
  for (int off = 16; off > 0; off >>= 1) scp += __shfl_xor(scp, off, 32);
  if (lane == 0) atomicAdd(&scacc, scp);    // ds_add_f32
#pragma unroll
  for (int q = 0; q < 8; ++q) atomicAdd(&yacc[lane + q * 32], bpart[q]);
  __syncthreads();
  y[(size_t)t * kD + tid] = scacc + yacc[tid];
}

// ---------------------------------------------------------------------------
// K4: out = y @ proj_w^T + proj_b  (4096x256x256 GEMM, fp32 WMMA)
// ---------------------------------------------------------------------------
__global__ __launch_bounds__(32) void k_proj(
    const float* __restrict__ y, const float* __restrict__ pw,
    const float* __restrict__ pb, float* __restrict__ out) {
  const int lane = threadIdx.x;
  const int t0 = blockIdx.x * 16;
  const int o0 = blockIdx.y * 16;
  const int mr = lane & 15;
  const int kb = (lane >> 4) * 2;
  const float* __restrict__ arow = y  + (size_t)(t0 + mr) * kD;
  const float* __restrict__ brow = pw + (size_t)(o0 + mr) * kD;
  v8f acc = {};
#pragma unroll 8
  for (int d0 = 0; d0 < kD; d0 += 4) {
    v2f a, b;
    a.x = arow[d0 + kb]; a.y = arow[d0 + kb + 1];
    b.x = brow[d0 + kb]; b.y = brow[d0 + kb + 1];
    acc = __builtin_amdgcn_wmma_f32_16x16x4_f32(false, a, false, b,
                                                (short)0, acc, false, false);
  }
  const float bias = pb[o0 + mr];
  const int half = (lane >> 4) * 8;
  float* __restrict__ dst = out + (size_t)(t0 + half) * kD + o0 + mr;
#pragma unroll
  for (int r = 0; r < 8; ++r) dst[(size_t)r * kD] = acc[r] + bias;
}

// ---------------------------------------------------------------------------
extern "C" void kernel_launch(void* const* d_in, const int* in_sizes, int n_in,
                              void* d_out, int out_size, void* d_ws,
                              size_t ws_size, hipStream_t stream) {
  const float*         x    = (const float*)d_in[0];
  const unsigned char* bits = (const unsigned char*)d_in[1]; // bool array
  const float*         keys = (const float*)d_in[2];
  const float*         ew   = (const float*)d_in[3];
  const float*         eb   = (const float*)d_in[4];
  const float*         pw   = (const float*)d_in[5];
  const float*         pb   = (const float*)d_in[6];
  float* out = (float*)d_out;

  // workspace layout (bytes)
  char* ws = (char*)d_ws;
  float* scores = (float*)ws;                                    // 16 MB
  size_t off = (size_t)kB * kH * kS * kNK * sizeof(float);
  int* eidx = (int*)(ws + off);                                  // 512 KB
  off += (size_t)kT * kTopK * sizeof(int);
  float* acoef = (float*)(ws + off);                             // 7.5 MB
  off += (size_t)kT * kTopK * kNN * sizeof(float);
  float* yv = (float*)(ws + off);                                // 4 MB

  k_score_gemm<<<dim3(kS / 16, kNK / 16, kB * kH), 32, 0, stream>>>(
      x, keys, scores);
  k_topk<<<kT, 32, 0, stream>>>(scores, bits, eidx, acoef);
  k_experts<<<kT, 256, 0, stream>>>(x, ew, eb, eidx, acoef, yv);
  k_proj<<<dim3(kT / 16, kD / 16), 32, 0, stream>>>(yv, pw, pb, out);
}